// scGAT_nodropout_40106404610257
// MI455X (gfx1250) — compile-verified
//
#include <hip/hip_runtime.h>
#include <math.h>

// ---------------------------------------------------------------------------
// scGAT (2-layer GATConv, fp32) for MI455X / gfx1250.
// GEMMs use native fp32 WMMA (v_wmma_f32_16x16x4_f32) - full precision,
// matrix-engine path. Edge softmax/scatter uses f32 global atomics.
// ---------------------------------------------------------------------------

typedef float v2f __attribute__((ext_vector_type(2)));
typedef float v8f __attribute__((ext_vector_type(8)));

__device__ __forceinline__ float lrelu(float v) { return v > 0.f ? v : 0.2f * v; }

// Sign-aware float atomic max via integer ordering trick (native atomics).
__device__ __forceinline__ void atomicMaxF(float* addr, float val) {
    if (val >= 0.f)
        atomicMax((int*)addr, __float_as_int(val));
    else
        atomicMin((unsigned int*)addr, (unsigned int)__float_as_int(val));
}

// ---------------------------- utility fills --------------------------------
__global__ void fill_kernel(float* __restrict__ p, float v, long long n) {
    long long i = blockIdx.x * (long long)blockDim.x + threadIdx.x;
    if (i < n) p[i] = v;
}

// ---------------------------- WMMA fp32 GEMM -------------------------------
// C[nrows x ND] = A[nrows x KD] * B[KD x ND], row-major, fp32.
// One wave per 16x16 tile of C; wave w in the block owns column tile w.
// A-matrix per-lane: K = j + 2*(lane>=16), M = lane&15      (ISA 7.12.2)
// B-matrix per-lane: K = j + 2*(lane>=16), N = lane&15
// C/D: VGPR r -> M = r + 8*(lane>=16), N = lane&15
template <int KD, int ND>
__global__ void wmma_gemm_f32(const float* __restrict__ A,
                              const float* __restrict__ B,
                              float* __restrict__ C,
                              int nrows) {
    const int lane = threadIdx.x & 31;
    const int wave = threadIdx.x >> 5;
    const int row0 = blockIdx.x << 4;
    const int col0 = wave << 4;
    const int m    = lane & 15;
    const int half = lane >> 4;
    const int kk   = half << 1;

    int arow = row0 + m;
    if (arow >= nrows) arow = nrows - 1;   // clamp (tail safety; N%16==0 here)
    const float* ap = A + (long long)arow * KD;

    v8f acc = {0.f, 0.f, 0.f, 0.f, 0.f, 0.f, 0.f, 0.f};

    #pragma unroll 8
    for (int k = 0; k < KD; k += 4) {
        v2f a, b;
        a.x = ap[k + kk];
        a.y = ap[k + kk + 1];
        b.x = B[(k + kk) * ND + col0 + m];
        b.y = B[(k + kk + 1) * ND + col0 + m];
        acc = __builtin_amdgcn_wmma_f32_16x16x4_f32(
            /*neg_a=*/false, a, /*neg_b=*/false, b,
            /*c_mod=*/(short)0, acc, /*reuse_a=*/false, /*reuse_b=*/false);
    }

    const int rbase = row0 + (half << 3);
    float* cp = C + (long long)rbase * ND + col0 + m;
    if (row0 + 16 <= nrows) {
        // Uniform fast path: whole tile in range, unpredicated stores.
        #pragma unroll
        for (int r = 0; r < 8; ++r) cp[(long long)r * ND] = acc[r];
    } else {
        #pragma unroll
        for (int r = 0; r < 8; ++r) {
            if (rbase + r < nrows) cp[(long long)r * ND] = acc[r];
        }
    }
}

// ------------------- per-node attention coefficients -----------------------
// AS[n,h] = <h_mat[n,h,:], a_src[h,:]>, AD likewise. One thread per (node,head).
__global__ void attn_coef(const float* __restrict__ Hm,
                          const float* __restrict__ a_s,
                          const float* __restrict__ a_d,
                          float* __restrict__ AS, float* __restrict__ AD,
                          int n, int C, int stride) {
    int t = blockIdx.x * blockDim.x + threadIdx.x;
    if (t >= n * 8) return;
    int node = t >> 3;
    int h    = t & 7;
    const float* hp = Hm + (long long)node * stride + h * C;
    const float* sv = a_s + h * C;
    const float* dv = a_d + h * C;
    float s = 0.f, d = 0.f;
    for (int c = 0; c < C; ++c) {
        float v = hp[c];
        s += v * sv[c];
        d += v * dv[c];
    }
    AS[t] = s;
    AD[t] = d;
}

// ------------------------- edge softmax passes -----------------------------
// Edge ids [0,E) come from edge_index (int64); ids [E,ET) are self-loops.
__global__ void edge_max8(const long long* __restrict__ ei, long long E, long long ET,
                          const float* __restrict__ AS, const float* __restrict__ AD,
                          float* __restrict__ Mx) {
    long long e = blockIdx.x * (long long)blockDim.x + threadIdx.x;
    if (e >= ET) return;
    int s = (e < E) ? (int)ei[e]     : (int)(e - E);
    int d = (e < E) ? (int)ei[E + e] : (int)(e - E);
    const float4* a4 = (const float4*)(AS + (long long)s * 8);
    const float4* b4 = (const float4*)(AD + (long long)d * 8);
    float4 a0 = a4[0], a1 = a4[1], b0 = b4[0], b1 = b4[1];
    float v[8] = {a0.x + b0.x, a0.y + b0.y, a0.z + b0.z, a0.w + b0.w,
                  a1.x + b1.x, a1.y + b1.y, a1.z + b1.z, a1.w + b1.w};
    float* mp = Mx + (long long)d * 8;
    #pragma unroll
    for (int h = 0; h < 8; ++h) atomicMaxF(mp + h, lrelu(v[h]));
}

__global__ void edge_sum8(const long long* __restrict__ ei, long long E, long long ET,
                          const float* __restrict__ AS, const float* __restrict__ AD,
                          const float* __restrict__ Mx, float* __restrict__ Sm) {
    long long e = blockIdx.x * (long long)blockDim.x + threadIdx.x;
    if (e >= ET) return;
    int s = (e < E) ? (int)ei[e]     : (int)(e - E);
    int d = (e < E) ? (int)ei[E + e] : (int)(e - E);
    const float4* a4 = (const float4*)(AS + (long long)s * 8);
    const float4* b4 = (const float4*)(AD + (long long)d * 8);
    float4 a0 = a4[0], a1 = a4[1], b0 = b4[0], b1 = b4[1];
    float v[8] = {a0.x + b0.x, a0.y + b0.y, a0.z + b0.z, a0.w + b0.w,
                  a1.x + b1.x, a1.y + b1.y, a1.z + b1.z, a1.w + b1.w};
    const float* mp = Mx + (long long)d * 8;
    float* sp = Sm + (long long)d * 8;
    #pragma unroll
    for (int h = 0; h < 8; ++h) atomicAdd(sp + h, expf(lrelu(v[h]) - mp[h]));
}

// ------------------------- message scatter pass ----------------------------
// One thread per (edge, head): recompute alpha, scatter C channels.
template <int C, int STRIDE>
__global__ void edge_scatter(const long long* __restrict__ ei, long long E, long long ET,
                             const float* __restrict__ AS, const float* __restrict__ AD,
                             const float* __restrict__ Mx, const float* __restrict__ Sm,
                             const float* __restrict__ Hsrc, float* __restrict__ Out) {
    long long t = blockIdx.x * (long long)blockDim.x + threadIdx.x;
    if (t >= ET * 8) return;
    long long e = t >> 3;
    int h = (int)(t & 7);
    int s = (e < E) ? (int)ei[e]     : (int)(e - E);
    int d = (e < E) ? (int)ei[E + e] : (int)(e - E);
    float logit = lrelu(AS[(long long)s * 8 + h] + AD[(long long)d * 8 + h]);
    float alpha = expf(logit - Mx[(long long)d * 8 + h]) / Sm[(long long)d * 8 + h];
    const float* hp = Hsrc + (long long)s * STRIDE + h * C;
    float* op       = Out  + (long long)d * STRIDE + h * C;
    #pragma unroll
    for (int c = 0; c < C; ++c) atomicAdd(op + c, hp[c] * alpha);
}

// ----------------------------- epilogues -----------------------------------
__global__ void bias_elu(float* __restrict__ O, const float* __restrict__ b, long long n) {
    long long i = blockIdx.x * (long long)blockDim.x + threadIdx.x;
    if (i >= n) return;
    float v = O[i] + b[i & 63];
    O[i] = v > 0.f ? v : expm1f(v);
}

__global__ void mean_bias_logsoftmax(const float* __restrict__ O2,
                                     const float* __restrict__ b2,
                                     float* __restrict__ out, int n) {
    int node = blockIdx.x * blockDim.x + threadIdx.x;
    if (node >= n) return;
    float acc[10];
    #pragma unroll
    for (int c = 0; c < 10; ++c) acc[c] = 0.f;
    const float* p = O2 + (long long)node * 80;
    #pragma unroll
    for (int h = 0; h < 8; ++h) {
        #pragma unroll
        for (int c = 0; c < 10; ++c) acc[c] += p[h * 10 + c];
    }
    float mx = -INFINITY;
    #pragma unroll
    for (int c = 0; c < 10; ++c) {
        acc[c] = acc[c] * 0.125f + b2[c];
        mx = fmaxf(mx, acc[c]);
    }
    float sum = 0.f;
    #pragma unroll
    for (int c = 0; c < 10; ++c) sum += expf(acc[c] - mx);
    float l = mx + logf(sum);
    #pragma unroll
    for (int c = 0; c < 10; ++c) out[node * 10 + c] = acc[c] - l;
}

// ---------------------------------------------------------------------------
extern "C" void kernel_launch(void* const* d_in, const int* in_sizes, int n_in,
                              void* d_out, int out_size, void* d_ws, size_t ws_size,
                              hipStream_t stream) {
    const float*     x      = (const float*)d_in[0];
    const long long* ei     = (const long long*)d_in[1];   // int64 edge_index [2,E]
    const float*     W1     = (const float*)d_in[2];
    const float*     a_src1 = (const float*)d_in[3];
    const float*     a_dst1 = (const float*)d_in[4];
    const float*     b1     = (const float*)d_in[5];
    const float*     W2     = (const float*)d_in[6];
    const float*     a_src2 = (const float*)d_in[7];
    const float*     a_dst2 = (const float*)d_in[8];
    const float*     b2     = (const float*)d_in[9];

    const int       N  = in_sizes[0] / 128;
    const long long E  = in_sizes[1] / 2;
    const long long ET = E + N;                 // + self-loops

    // Workspace layout (floats). Hbuf region reused: H1 (64N) then H2 (80N).
    float* ws   = (float*)d_ws;
    float* Hbuf = ws;                              // [0, 80N)
    float* O1   = ws + (long long)N * 80;          // [80N, 144N)
    float* O2   = ws + (long long)N * 144;         // [144N, 224N)
    float* AS   = ws + (long long)N * 224;         // [224N, 232N)
    float* AD   = ws + (long long)N * 232;         // [232N, 240N)
    float* Mx   = ws + (long long)N * 240;         // [240N, 248N)
    float* Sm   = ws + (long long)N * 248;         // [248N, 256N)  total ~102 MB

    const int TB = 256;
    auto nb = [](long long n, int tb) { return (unsigned)((n + tb - 1) / tb); };
    const unsigned gtiles = (unsigned)((N + 15) / 16);

    // ----------------------------- layer 1 --------------------------------
    fill_kernel<<<nb(8LL * N, TB), TB, 0, stream>>>(Mx, -INFINITY, 8LL * N);
    fill_kernel<<<nb(8LL * N, TB), TB, 0, stream>>>(Sm, 0.f, 8LL * N);
    fill_kernel<<<nb(64LL * N, TB), TB, 0, stream>>>(O1, 0.f, 64LL * N);

    wmma_gemm_f32<128, 64><<<gtiles, 128, 0, stream>>>(x, W1, Hbuf, N);
    attn_coef<<<nb(8LL * N, TB), TB, 0, stream>>>(Hbuf, a_src1, a_dst1, AS, AD, N, 8, 64);

    edge_max8<<<nb(ET, TB), TB, 0, stream>>>(ei, E, ET, AS, AD, Mx);
    edge_sum8<<<nb(ET, TB), TB, 0, stream>>>(ei, E, ET, AS, AD, Mx, Sm);
    edge_scatter<8, 64><<<nb(ET * 8, TB), TB, 0, stream>>>(ei, E, ET, AS, AD, Mx, Sm, Hbuf, O1);

    bias_elu<<<nb(64LL * N, TB), TB, 0, stream>>>(O1, b1, 64LL * N);

    // ----------------------------- layer 2 --------------------------------
    fill_kernel<<<nb(8LL * N, TB), TB, 0, stream>>>(Mx, -INFINITY, 8LL * N);
    fill_kernel<<<nb(8LL * N, TB), TB, 0, stream>>>(Sm, 0.f, 8LL * N);
    fill_kernel<<<nb(80LL * N, TB), TB, 0, stream>>>(O2, 0.f, 80LL * N);

    wmma_gemm_f32<64, 80><<<gtiles, 160, 0, stream>>>(O1, W2, Hbuf, N);
    attn_coef<<<nb(8LL * N, TB), TB, 0, stream>>>(Hbuf, a_src2, a_dst2, AS, AD, N, 10, 80);

    edge_max8<<<nb(ET, TB), TB, 0, stream>>>(ei, E, ET, AS, AD, Mx);
    edge_sum8<<<nb(ET, TB), TB, 0, stream>>>(ei, E, ET, AS, AD, Mx, Sm);
    edge_scatter<10, 80><<<nb(ET * 8, TB), TB, 0, stream>>>(ei, E, ET, AS, AD, Mx, Sm, Hbuf, O2);

    mean_bias_logsoftmax<<<nb(N, TB), TB, 0, stream>>>(O2, b2, (float*)d_out, N);

    (void)n_in; (void)out_size; (void)ws_size;
}